// M13_72164040508123
// MI455X (gfx1250) — compile-verified
//
#include <hip/hip_runtime.h>

// ---------------------------------------------------------------------------
// CDNA5 (gfx1250) GINEConv GNN forward. fp32 WMMA 16x16x4 for all GEMMs.
// ---------------------------------------------------------------------------

typedef float v2f __attribute__((ext_vector_type(2)));
typedef float v8f __attribute__((ext_vector_type(8)));

static __device__ __forceinline__ v8f wmma4(v2f a, v2f b, v8f c) {
  // D = A(16x4 f32) x B(4x16 f32) + C(16x16 f32)
  return __builtin_amdgcn_wmma_f32_16x16x4_f32(false, a, false, b, (short)0, c,
                                               false, false);
}

static __device__ __forceinline__ float leaky(float v) {
  return v >= 0.0f ? v : 0.01f * v;
}

static __device__ __forceinline__ void atomAddF(float* p, float v) {
  __hip_atomic_fetch_add(p, v, __ATOMIC_RELAXED, __HIP_MEMORY_SCOPE_AGENT);
}

// ---------------------------------------------------------------------------
// Edge kernel: one wave per 16-edge tile.
//   msg = relu(x[src] + edge_attr @ We + be);  agg[dst] += msg (atomic)
// edge_attr tile: A = [16 edges x 16 feats], We: [16 x 64], 4 N-blocks of 16.
// ---------------------------------------------------------------------------
__global__ __launch_bounds__(256) void edge_kernel(
    const float* __restrict__ x, const float* __restrict__ ea,
    const int* __restrict__ ei, const float* __restrict__ We,
    const float* __restrict__ be, float* __restrict__ agg, int E) {
  const int lane = threadIdx.x & 31;
  const int wave = threadIdx.x >> 5;
  const long tile = (long)blockIdx.x * 8 + wave;
  const long ebase = tile * 16;
  if (ebase >= (long)E) return;
  const int half = lane >> 4;  // 0: K 0..1, rows M; 1: K 2..3, rows M+8 in D
  const int l16 = lane & 15;
  const int koff = half * 2;

  // A fragments for the 4 K-steps (K = 16 total, 4 per WMMA)
  v2f a[4];
#pragma unroll
  for (int ks = 0; ks < 4; ++ks) {
    const float* row = ea + (ebase + l16) * 16 + ks * 4 + koff;
    a[ks].x = row[0];
    a[ks].y = row[1];
  }
  const int* __restrict__ srcp = ei;
  const int* __restrict__ dstp = ei + E;

#pragma unroll
  for (int nb = 0; nb < 4; ++nb) {
    v8f acc = {};
#pragma unroll
    for (int ks = 0; ks < 4; ++ks) {
      const int k0 = ks * 4 + koff;
      v2f b;
      b.x = We[(k0 + 0) * 64 + nb * 16 + l16];
      b.y = We[(k0 + 1) * 64 + nb * 16 + l16];
      acc = wmma4(a[ks], b, acc);
    }
    const int col = nb * 16 + l16;
    const float bec = be[col];
#pragma unroll
    for (int r = 0; r < 8; ++r) {
      const long e = ebase + r + half * 8;
      const int s = srcp[e];
      const int d = dstp[e];
      float msg = acc[r] + bec + x[(long)s * 64 + col];
      msg = msg > 0.0f ? msg : 0.0f;
      atomAddF(&agg[(long)d * 64 + col], msg);
    }
  }
}

// ---------------------------------------------------------------------------
// Generic fused WMMA GEMM: out = f(A) @ W (+bias) (+out), optional BN stats.
//   ADD : A_eff = (1+*epsPtr)*A + Aadd          (GINE node combine)
//   AFF : A_eff = leaky(A*inScale[k]+inShift[k]) (fused BN+LeakyReLU on input)
//   STATS: atomically accumulate per-column sum / sumsq of the stored output.
// One wave per 16-row tile; compile-time K/NOUT -> fully unrolled, regs only.
// ---------------------------------------------------------------------------
template <int K, int NOUT, bool ADD, bool AFF, bool BIAS, bool ACCUM, bool STATS>
__global__ __launch_bounds__(256) void gemm_kernel(
    const float* __restrict__ A, const float* __restrict__ Aadd,
    const float* __restrict__ epsPtr, const float* __restrict__ inScale,
    const float* __restrict__ inShift, const float* __restrict__ W,
    const float* __restrict__ bias, float* __restrict__ out,
    float* __restrict__ stats, int Nrows) {
  constexpr int KS = K / 4;
  constexpr int NB = NOUT / 16;
  const int lane = threadIdx.x & 31;
  const int wave = threadIdx.x >> 5;
  const long tile = (long)blockIdx.x * 8 + wave;
  const long rbase = tile * 16;
  if (rbase >= (long)Nrows) return;
  const int half = lane >> 4;
  const int l16 = lane & 15;
  const int koff = half * 2;

  float alpha = 1.0f;
  if (ADD) alpha = 1.0f + *epsPtr;

  v8f acc[NB];
  v8f vzero = {};
#pragma unroll
  for (int nb = 0; nb < NB; ++nb) acc[nb] = vzero;

  const long ro = (rbase + l16) * (long)K;
#pragma unroll
  for (int ks = 0; ks < KS; ++ks) {
    const int k0 = ks * 4 + koff;
    float a0 = A[ro + k0];
    float a1 = A[ro + k0 + 1];
    if (ADD) {
      a0 = alpha * a0 + Aadd[ro + k0];
      a1 = alpha * a1 + Aadd[ro + k0 + 1];
    }
    if (AFF) {
      a0 = leaky(a0 * inScale[k0] + inShift[k0]);
      a1 = leaky(a1 * inScale[k0 + 1] + inShift[k0 + 1]);
    }
    v2f a;
    a.x = a0;
    a.y = a1;
#pragma unroll
    for (int nb = 0; nb < NB; ++nb) {
      v2f b;
      b.x = W[(k0 + 0) * NOUT + nb * 16 + l16];
      b.y = W[(k0 + 1) * NOUT + nb * 16 + l16];
      acc[nb] = wmma4(a, b, acc[nb]);
    }
  }

#pragma unroll
  for (int nb = 0; nb < NB; ++nb) {
    const int col = nb * 16 + l16;
    const float bv = BIAS ? bias[col] : 0.0f;
    float s1 = 0.0f, s2 = 0.0f;
#pragma unroll
    for (int r = 0; r < 8; ++r) {
      const long row = rbase + r + half * 8;
      const long o = row * (long)NOUT + col;
      float v = acc[nb][r] + bv;
      if (ACCUM) v += out[o];
      out[o] = v;
      if (STATS) {
        s1 += v;
        s2 += v * v;
      }
    }
    if (STATS) {
      s1 += __shfl_xor(s1, 16);
      s2 += __shfl_xor(s2, 16);
      if (half == 0) {
        atomAddF(&stats[col], s1);
        atomAddF(&stats[NOUT + col], s2);
      }
    }
  }
}

// stats(sum,sumsq) -> fused BN scale/shift (training-mode, biased var)
__global__ void finalize_kernel(const float* __restrict__ stats,
                                const float* __restrict__ g,
                                const float* __restrict__ b, float invN,
                                float* __restrict__ scale,
                                float* __restrict__ shift, int C) {
  const int c = blockIdx.x * blockDim.x + threadIdx.x;
  if (c < C) {
    const float m = stats[c] * invN;
    const float var = stats[C + c] * invN - m * m;
    const float is = rsqrtf(var + 1e-5f);
    const float sc = g[c] * is;
    scale[c] = sc;
    shift[c] = b[c] - sc * m;
  }
}

// in-place x = leaky(scale[c]*x + shift[c]), C power-of-two
__global__ __launch_bounds__(256) void bnleaky_kernel(
    float* __restrict__ x, const float* __restrict__ scale,
    const float* __restrict__ shift, long n, int Cmask) {
  const long i = (long)blockIdx.x * blockDim.x + threadIdx.x;
  if (i < n) {
    const int c = (int)(i & Cmask);
    x[i] = leaky(x[i] * scale[c] + shift[c]);
  }
}

// out[i] = leaky(bn(t2[i,:])) . Wf2 + bf2
__global__ __launch_bounds__(256) void out_kernel(
    const float* __restrict__ t2, const float* __restrict__ scale,
    const float* __restrict__ shift, const float* __restrict__ Wf2,
    const float* __restrict__ bf2, float* __restrict__ out, int N) {
  const int i = blockIdx.x * blockDim.x + threadIdx.x;
  if (i >= N) return;
  const float* row = t2 + (long)i * 128;
  float acc = 0.0f;
#pragma unroll
  for (int k = 0; k < 128; ++k) {
    acc += leaky(row[k] * scale[k] + shift[k]) * Wf2[k];
  }
  out[i] = acc + bf2[0];
}

// ---------------------------------------------------------------------------
extern "C" void kernel_launch(void* const* d_in, const int* in_sizes, int n_in,
                              void* d_out, int out_size, void* d_ws,
                              size_t ws_size, hipStream_t stream) {
  (void)n_in;
  (void)out_size;
  (void)ws_size;
  const float* x0 = (const float*)d_in[0];
  const float* ea = (const float*)d_in[1];
  const float* molx = (const float*)d_in[2];
  const int* ei = (const int*)d_in[3];
  const float* We = (const float*)d_in[4];
  const float* be = (const float*)d_in[5];
  const float* eps = (const float*)d_in[6];
  const float* Wa = (const float*)d_in[7];
  const float* ba = (const float*)d_in[8];
  const float* ga = (const float*)d_in[9];
  const float* bga = (const float*)d_in[10];
  const float* Wb = (const float*)d_in[11];
  const float* bb = (const float*)d_in[12];
  const float* go = (const float*)d_in[13];
  const float* bo = (const float*)d_in[14];
  const float* Wm1 = (const float*)d_in[15];
  const float* bm1 = (const float*)d_in[16];
  const float* gm = (const float*)d_in[17];
  const float* bgm = (const float*)d_in[18];
  const float* Wm2 = (const float*)d_in[19];
  const float* bm2 = (const float*)d_in[20];
  const float* Wf1 = (const float*)d_in[21];
  const float* bf1 = (const float*)d_in[22];
  const float* gf = (const float*)d_in[23];
  const float* bgf = (const float*)d_in[24];
  const float* Wf2 = (const float*)d_in[25];
  const float* bf2 = (const float*)d_in[26];
  float* out = (float*)d_out;

  const int N = in_sizes[0] / 64;  // 100000
  const int E = in_sizes[1] / 16;  // 1600000
  const float invN = 1.0f / (float)N;

  float* ws = (float*)d_ws;
  float* bufX = ws;                          // [N,64] current node features
  float* bufAgg = ws + (size_t)N * 64;       // [N,64] edge aggregation
  float* bufT = ws + (size_t)2 * N * 64;     // [N,64] inner hidden
  float* bufXm = ws + (size_t)3 * N * 64;    // [N,64] mol features
  float* bufT2 = bufAgg;                     // [N,128] final hidden (alias agg+t)
  float* stats = ws + (size_t)4 * N * 64;    // 256: sum | sumsq
  float* scale = stats + 256;                // 128
  float* shift = scale + 128;                // 128

  const int edgeBlocks = (E / 16 + 7) / 8;
  const int nodeBlocks = ((N + 15) / 16 + 7) / 8;
  const size_t statsB = 256 * sizeof(float);

  // ---- GINEConv layers ----
  for (int l = 0; l < 3; ++l) {
    const float* xin = (l == 0) ? x0 : bufX;
    hipMemsetAsync(bufAgg, 0, (size_t)N * 64 * sizeof(float), stream);
    edge_kernel<<<edgeBlocks, 256, 0, stream>>>(xin, ea, ei, We + l * 16 * 64,
                                                be + l * 64, bufAgg, E);
    // t = ((1+eps)*x + agg) @ Wa + ba, accumulate BN stats
    hipMemsetAsync(stats, 0, statsB, stream);
    gemm_kernel<64, 64, true, false, true, false, true>
        <<<nodeBlocks, 256, 0, stream>>>(xin, bufAgg, eps + l, nullptr, nullptr,
                                         Wa + l * 64 * 64, ba + l * 64, bufT,
                                         stats, N);
    finalize_kernel<<<1, 64, 0, stream>>>(stats, ga + l * 64, bga + l * 64,
                                          invN, scale, shift, 64);
    if (l < 2) {
      // x = leaky(bn(t)) @ Wb + bb, accumulate outer BN stats
      hipMemsetAsync(stats, 0, statsB, stream);
      gemm_kernel<64, 64, false, true, true, false, true>
          <<<nodeBlocks, 256, 0, stream>>>(bufT, nullptr, nullptr, scale, shift,
                                           Wb + l * 64 * 64, bb + l * 64, bufX,
                                           stats, N);
      finalize_kernel<<<1, 64, 0, stream>>>(stats, go + l * 64, bo + l * 64,
                                            invN, scale, shift, 64);
      bnleaky_kernel<<<(int)(((long)N * 64 + 255) / 256), 256, 0, stream>>>(
          bufX, scale, shift, (long)N * 64, 63);
    } else {
      gemm_kernel<64, 64, false, true, true, false, false>
          <<<nodeBlocks, 256, 0, stream>>>(bufT, nullptr, nullptr, scale, shift,
                                           Wb + l * 64 * 64, bb + l * 64, bufX,
                                           nullptr, N);
    }
  }

  // ---- mol MLP: xm = leaky(bn(mol_x@Wm1+bm1)) @ Wm2 + bm2 ----
  hipMemsetAsync(stats, 0, statsB, stream);
  gemm_kernel<32, 64, false, false, true, false, true>
      <<<nodeBlocks, 256, 0, stream>>>(molx, nullptr, nullptr, nullptr, nullptr,
                                       Wm1, bm1, bufT, stats, N);
  finalize_kernel<<<1, 64, 0, stream>>>(stats, gm, bgm, invN, scale, shift, 64);
  gemm_kernel<64, 64, false, true, true, false, false>
      <<<nodeBlocks, 256, 0, stream>>>(bufT, nullptr, nullptr, scale, shift,
                                       Wm2, bm2, bufXm, nullptr, N);

  // ---- final MLP: split concat GEMM ----
  gemm_kernel<64, 128, false, false, false, false, false>
      <<<nodeBlocks, 256, 0, stream>>>(bufX, nullptr, nullptr, nullptr, nullptr,
                                       Wf1, nullptr, bufT2, nullptr, N);
  hipMemsetAsync(stats, 0, statsB, stream);
  gemm_kernel<64, 128, false, false, true, true, true>
      <<<nodeBlocks, 256, 0, stream>>>(bufXm, nullptr, nullptr, nullptr,
                                       nullptr, Wf1 + 64 * 128, bf1, bufT2,
                                       stats, N);
  finalize_kernel<<<1, 128, 0, stream>>>(stats, gf, bgf, invN, scale, shift,
                                         128);
  out_kernel<<<(N + 255) / 256, 256, 0, stream>>>(bufT2, scale, shift, Wf2, bf2,
                                                  out, N);
}